// EHimp_78116865180058
// MI455X (gfx1250) — compile-verified
//
#include <hip/hip_runtime.h>
#include <hip/hip_bf16.h>

// ---------------------------------------------------------------------------
// Types for CDNA5 WMMA (fp32 16x16x4): A/B = 2 floats/lane, C/D = 8 floats/lane
// ---------------------------------------------------------------------------
typedef float v2f __attribute__((ext_vector_type(2)));
typedef float v8f __attribute__((ext_vector_type(8)));

#define HDIM 128

// ---------------------------------------------------------------------------
// Generic fp32 WMMA GEMM:  Out[M,Nout] = A @ W + bias
//   A[m][k] = (epsP ? (1+*epsP) : 1) * X[m][k] + (Y ? Y[m][k] : 0)
// Block = 256 threads (8 waves). Block tile = 32 rows x 128 cols.
// Wave w: row-tile (w&1)*16, col base (w>>1)*32 (two 16-col tiles).
// K in {128,256}. M multiple of 32, Nout multiple of 128.
// ---------------------------------------------------------------------------
__global__ __launch_bounds__(256) void gemm_wmma_kernel(
    const float* __restrict__ X, const float* __restrict__ Y,
    const float* __restrict__ epsP, const float* __restrict__ W,
    const float* __restrict__ bias, float* __restrict__ Out,
    int M, int K, int Nout)
{
    (void)M;
    __shared__ float As[32 * 258];           // 32 rows x (K+2) stride, K<=256
    const int LDA = K + 2;
    const int tid  = threadIdx.x;
    const int lane = tid & 31;
    const int wave = tid >> 5;
    const int row0 = blockIdx.x * 32;
    const int colB = blockIdx.y * 128;

    const float c0 = epsP ? (1.0f + epsP[0]) : 1.0f;

    // ---- cooperative load of A tile (32 x K) into LDS, coalesced ----
    {
        const int m  = tid >> 3;             // 0..31
        const int kc = (tid & 7) * 4;        // 0,4,..,28
        const float* xrow = X + (size_t)(row0 + m) * (size_t)K;
        float* as = &As[m * LDA];
        if (Y) {
            const float* yrow = Y + (size_t)(row0 + m) * (size_t)K;
            for (int k = kc; k < K; k += 32) {
                float4 a = *(const float4*)(xrow + k);
                float4 b = *(const float4*)(yrow + k);
                as[k + 0] = c0 * a.x + b.x;
                as[k + 1] = c0 * a.y + b.y;
                as[k + 2] = c0 * a.z + b.z;
                as[k + 3] = c0 * a.w + b.w;
            }
        } else {
            for (int k = kc; k < K; k += 32) {
                float4 a = *(const float4*)(xrow + k);
                as[k + 0] = c0 * a.x;
                as[k + 1] = c0 * a.y;
                as[k + 2] = c0 * a.z;
                as[k + 3] = c0 * a.w;
            }
        }
    }
    __syncthreads();

    const int rowTile = (wave & 1) * 16;             // 0 or 16 (within block)
    const int colAbs  = colB + (wave >> 1) * 32;     // absolute col of first tile
    const int halfsel = lane >> 4;                   // 0: lanes0-15, 1: lanes16-31
    const int ln      = lane & 15;

    v8f acc0 = {};
    v8f acc1 = {};

    const float* arow = &As[(rowTile + ln) * LDA];
    for (int k = 0; k < K; k += 4) {
        const int ka = k + halfsel * 2;
        // A fragment: lane<16 -> A[m][k..k+1], lane>=16 -> A[m][k+2..k+3]
        v2f a;
        a.x = arow[ka];
        a.y = arow[ka + 1];
        // B fragments: lane<16 -> W[k..k+1][n], lane>=16 -> W[k+2..k+3][n]
        const float* wp0 = W + (size_t)ka * (size_t)Nout + colAbs + ln;
        v2f b0, b1;
        b0.x = wp0[0];
        b0.y = wp0[Nout];
        b1.x = wp0[16];
        b1.y = wp0[Nout + 16];
        acc0 = __builtin_amdgcn_wmma_f32_16x16x4_f32(false, a, false, b0,
                                                     (short)0, acc0, false, false);
        acc1 = __builtin_amdgcn_wmma_f32_16x16x4_f32(false, a, false, b1,
                                                     (short)0, acc1, false, false);
    }

    const float bv0 = bias ? bias[colAbs + ln]      : 0.0f;
    const float bv1 = bias ? bias[colAbs + 16 + ln] : 0.0f;
    const int rbase = row0 + rowTile + halfsel * 8;
#pragma unroll
    for (int r = 0; r < 8; ++r) {
        float* o = Out + (size_t)(rbase + r) * (size_t)Nout + colAbs + ln;
        o[0]  = acc0[r] + bv0;
        o[16] = acc1[r] + bv1;
    }
}

// ---------------------------------------------------------------------------
// Encoders
// ---------------------------------------------------------------------------
__global__ __launch_bounds__(HDIM) void atom_encode_kernel(
    const int* __restrict__ nf, const float* __restrict__ emb, float* __restrict__ x)
{
    const int i = blockIdx.x, h = threadIdx.x;
    float s = 0.0f;
#pragma unroll
    for (int k = 0; k < 9; ++k) {
        const int v = nf[i * 9 + k];
        s += emb[((size_t)k * 100 + v) * HDIM + h];
    }
    x[(size_t)i * HDIM + h] = s;
}

__global__ __launch_bounds__(HDIM) void rg_encode_kernel(
    const int* __restrict__ feat, const float* __restrict__ emb, float* __restrict__ rg0)
{
    const int i = blockIdx.x, h = threadIdx.x;
    rg0[(size_t)i * HDIM + h] = emb[(size_t)feat[i] * HDIM + h];
}

// ---------------------------------------------------------------------------
// GINEConv message: aggr[dst] += relu(x[src] + sum_k bond_emb[k][ef[e,k]])
// ---------------------------------------------------------------------------
__global__ __launch_bounds__(HDIM) void edge_msg_kernel(
    const float* __restrict__ x, const int* __restrict__ ef,
    const int* __restrict__ src, const int* __restrict__ dst,
    const float* __restrict__ bemb, float* __restrict__ aggr)
{
    const int e = blockIdx.x, h = threadIdx.x;
    const int s = src[e], d = dst[e];
    float m = x[(size_t)s * HDIM + h];
#pragma unroll
    for (int k = 0; k < 3; ++k)
        m += bemb[((size_t)k * 100 + ef[e * 3 + k]) * HDIM + h];
    m = fmaxf(m, 0.0f);
    atomicAdd(&aggr[(size_t)d * HDIM + h], m);
}

// ---------------------------------------------------------------------------
// Reduced-graph GIN aggregation: aggr[rdst] += rg[rsrc]
// ---------------------------------------------------------------------------
__global__ __launch_bounds__(HDIM) void rg_aggr_kernel(
    const float* __restrict__ rg, const int* __restrict__ rsrc,
    const int* __restrict__ rdst, float* __restrict__ aggr)
{
    const int e = blockIdx.x, h = threadIdx.x;
    atomicAdd(&aggr[(size_t)rdst[e] * HDIM + h], rg[(size_t)rsrc[e] * HDIM + h]);
}

// ---------------------------------------------------------------------------
// scatter-mean pass 1: sums[dst[i]] += data[srcIdx ? srcIdx[i] : i]; counts++
// ---------------------------------------------------------------------------
__global__ __launch_bounds__(HDIM) void scatter_add_kernel(
    const float* __restrict__ data, const int* __restrict__ srcIdx,
    const int* __restrict__ dstIdx, float* __restrict__ sums,
    float* __restrict__ counts)
{
    const int i = blockIdx.x, h = threadIdx.x;
    const int s = srcIdx ? srcIdx[i] : i;
    const int d = dstIdx[i];
    atomicAdd(&sums[(size_t)d * HDIM + h], data[(size_t)s * HDIM + h]);
    if (h == 0) atomicAdd(&counts[d], 1.0f);
}

__global__ __launch_bounds__(HDIM) void div_counts_kernel(
    float* __restrict__ sums, const float* __restrict__ counts)
{
    const int i = blockIdx.x, h = threadIdx.x;
    sums[(size_t)i * HDIM + h] /= fmaxf(counts[i], 1.0f);
}

// ---------------------------------------------------------------------------
// BatchNorm (training mode, biased var): column stats then apply + ReLU
// ---------------------------------------------------------------------------
__global__ __launch_bounds__(HDIM) void bn_stats_kernel(
    const float* __restrict__ h, int M, int C, int rowsPerBlock,
    double* __restrict__ stats)
{
    const int col = blockIdx.x * HDIM + threadIdx.x;
    int r0 = blockIdx.y * rowsPerBlock;
    int r1 = r0 + rowsPerBlock;
    if (r1 > M) r1 = M;
    float s = 0.0f, ss = 0.0f;
    for (int r = r0; r < r1; ++r) {
        const float v = h[(size_t)r * C + col];
        s += v;
        ss += v * v;
    }
    atomicAdd(&stats[col], (double)s);
    atomicAdd(&stats[C + col], (double)ss);
}

__global__ __launch_bounds__(256) void bn_relu_kernel(
    float* __restrict__ h, int M, int C,
    const double* __restrict__ stats,
    const float* __restrict__ g, const float* __restrict__ b)
{
    const size_t idx = (size_t)blockIdx.x * 256 + threadIdx.x;
    if (idx >= (size_t)M * (size_t)C) return;
    const int col = (int)(idx & (size_t)(C - 1));
    const double mu  = stats[col] / (double)M;
    const double var = stats[C + col] / (double)M - mu * mu;
    const float inv  = rsqrtf((float)var + 1e-5f);
    const float v = (h[idx] - (float)mu) * inv * g[col] + b[col];
    h[idx] = fmaxf(v, 0.0f);
}

// out = a + relu(b)  (elementwise; out may alias a)
__global__ __launch_bounds__(256) void add_relu_kernel(
    const float* __restrict__ a, const float* __restrict__ b,
    float* __restrict__ out, size_t total)
{
    const size_t i = (size_t)blockIdx.x * 256 + threadIdx.x;
    if (i >= total) return;
    out[i] = a[i] + fmaxf(b[i], 0.0f);
}

// ---------------------------------------------------------------------------
// Offsets of rg_num_atoms (exclusive prefix sum), two-pass, G<=16384
// ---------------------------------------------------------------------------
__global__ __launch_bounds__(256) void partial_sum_kernel(
    const int* __restrict__ cnt, int* __restrict__ partials, int G)
{
    __shared__ int red[256];
    const int t = threadIdx.x;
    const int i = blockIdx.x * 256 + t;
    red[t] = (i < G) ? cnt[i] : 0;
    __syncthreads();
    for (int s = 128; s > 0; s >>= 1) {
        if (t < s) red[t] += red[t + s];
        __syncthreads();
    }
    if (t == 0) partials[blockIdx.x] = red[0];
}

__global__ __launch_bounds__(256) void offsets_kernel(
    const int* __restrict__ cnt, const int* __restrict__ partials,
    int* __restrict__ offsets, int G)
{
    const int g = blockIdx.x * 256 + threadIdx.x;
    if (g >= G) return;
    const int b = g >> 8;
    int off = 0;
    for (int i = 0; i < b; ++i) off += partials[i];
    for (int i = b << 8; i < g; ++i) off += cnt[i];
    offsets[g] = off;
}

// mean of rg0 rows [off[g], off[g]+cnt[g]) per graph
__global__ __launch_bounds__(HDIM) void tree_pool_kernel(
    const float* __restrict__ rg0, const int* __restrict__ offsets,
    const int* __restrict__ cnts, float* __restrict__ out)
{
    const int g = blockIdx.x, h = threadIdx.x;
    const int o = offsets[g], c = cnts[g];
    float s = 0.0f;
    for (int j = 0; j < c; ++j) s += rg0[(size_t)(o + j) * HDIM + h];
    out[(size_t)g * HDIM + h] = s / fmaxf((float)c, 1.0f);
}

// out[g] = relu(xg[g]+rgp[g]) . lin_W + lin_b   (OUT = 1)
__global__ __launch_bounds__(HDIM) void final_head_kernel(
    const float* __restrict__ xg, const float* __restrict__ rgp,
    const float* __restrict__ linW, const float* __restrict__ linb,
    float* __restrict__ out)
{
    __shared__ float red[HDIM];
    const int g = blockIdx.x, t = threadIdx.x;
    red[t] = fmaxf(xg[(size_t)g * HDIM + t] + rgp[(size_t)g * HDIM + t], 0.0f) * linW[t];
    __syncthreads();
    for (int s = 64; s > 0; s >>= 1) {
        if (t < s) red[t] += red[t + s];
        __syncthreads();
    }
    if (t == 0) out[g] = red[0] + linb[0];
}

// ---------------------------------------------------------------------------
// Host launcher
// ---------------------------------------------------------------------------
extern "C" void kernel_launch(void* const* d_in, const int* in_sizes, int n_in,
                              void* d_out, int out_size, void* d_ws, size_t ws_size,
                              hipStream_t stream)
{
    (void)n_in; (void)out_size; (void)ws_size;
    const int H = HDIM;
    const int N   = in_sizes[0] / 9;
    const int E   = in_sizes[2] / 2;
    const int NR  = in_sizes[4];
    const int ERG = in_sizes[5] / 2;
    const int G   = in_sizes[7];
    const int L   = in_sizes[11];

    const int*   node_feat  = (const int*)d_in[0];
    const int*   edge_feat  = (const int*)d_in[1];
    const int*   edge_index = (const int*)d_in[2];
    const int*   batch      = (const int*)d_in[3];
    const int*   rg_feat    = (const int*)d_in[4];
    const int*   rg_eidx    = (const int*)d_in[5];
    const int*   mapping    = (const int*)d_in[6];
    const int*   rg_natoms  = (const int*)d_in[7];
    const float* atom_emb   = (const float*)d_in[8];
    const float* bond_emb   = (const float*)d_in[9];
    const float* rg_emb     = (const float*)d_in[10];
    const float* atom_eps   = (const float*)d_in[11];
    const float* atom_W1    = (const float*)d_in[12];
    const float* atom_b1    = (const float*)d_in[13];
    const float* atom_bn1g  = (const float*)d_in[14];
    const float* atom_bn1b  = (const float*)d_in[15];
    const float* atom_W2    = (const float*)d_in[16];
    const float* atom_b2    = (const float*)d_in[17];
    const float* atom_bng   = (const float*)d_in[18];
    const float* atom_bnb   = (const float*)d_in[19];
    const float* rg_eps     = (const float*)d_in[20];
    const float* rg_W1      = (const float*)d_in[21];
    const float* rg_b1      = (const float*)d_in[22];
    const float* rg_bn1g    = (const float*)d_in[23];
    const float* rg_bn1b    = (const float*)d_in[24];
    const float* rg_W2      = (const float*)d_in[25];
    const float* rg_b2      = (const float*)d_in[26];
    const float* rg_bng     = (const float*)d_in[27];
    const float* rg_bnb     = (const float*)d_in[28];
    const float* raw2rg_W   = (const float*)d_in[29];
    const float* raw2rg_b   = (const float*)d_in[30];
    const float* rg2raw_W   = (const float*)d_in[31];
    const float* rg2raw_b   = (const float*)d_in[32];
    const float* a_lin_W    = (const float*)d_in[33];
    const float* a_lin_b    = (const float*)d_in[34];
    const float* r_lin_W    = (const float*)d_in[35];
    const float* r_lin_b    = (const float*)d_in[36];
    const float* lin_W      = (const float*)d_in[37];
    const float* lin_b      = (const float*)d_in[38];

    // ---- workspace carve ----
    char* p = (char*)d_ws;
    auto carve = [&](size_t bytes) -> void* {
        void* r = (void*)p;
        p += (bytes + 255) & ~(size_t)255;
        return r;
    };
    float*  x       = (float*)carve((size_t)N * H * 4);
    float*  bufN1   = (float*)carve((size_t)N * H * 4);        // aggr / rg2raw sums
    float*  bufN2   = (float*)carve((size_t)N * 2 * H * 4);    // h1 / rg h1 / rg2raw out
    float*  rg0     = (float*)carve((size_t)NR * H * 4);
    float*  rg      = (float*)carve((size_t)NR * H * 4);
    float*  rgA     = (float*)carve((size_t)NR * H * 4);       // raw2rg sums / rg aggr
    float*  rgB     = (float*)carve((size_t)NR * H * 4);       // raw2rg gemm out
    float*  countsN = (float*)carve((size_t)N * 4);
    float*  countsR = (float*)carve((size_t)NR * 4);
    float*  countsG = (float*)carve((size_t)G * 4);
    double* stats   = (double*)carve(2 * 256 * 8);
    float*  poolX   = (float*)carve((size_t)G * H * 4);
    float*  xg      = (float*)carve((size_t)G * H * 4);
    float*  poolR   = (float*)carve((size_t)G * H * 4);
    float*  rgp     = (float*)carve((size_t)G * H * 4);
    int*    partials= (int*)carve(64 * 4);
    int*    offsets = (int*)carve((size_t)G * 4);

    const int* src  = edge_index;         // edge_index[0]
    const int* dst  = edge_index + E;     // edge_index[1]
    const int* rsrc = rg_eidx;
    const int* rdst = rg_eidx + ERG;
    const int* mrow = mapping;            // mapping[0]
    const int* mcol = mapping + N;        // mapping[1]

    const int RPB = 1024;                 // rows per bn_stats block
    auto gyN  = (unsigned)((N  + RPB - 1) / RPB);
    auto gyNR = (unsigned)((NR + RPB - 1) / RPB);

    // ---- encoders + tree offsets ----
    atom_encode_kernel<<<N, H, 0, stream>>>(node_feat, atom_emb, x);
    rg_encode_kernel<<<NR, H, 0, stream>>>(rg_feat, rg_emb, rg0);
    partial_sum_kernel<<<(G + 255) / 256, 256, 0, stream>>>(rg_natoms, partials, G);
    offsets_kernel<<<(G + 255) / 256, 256, 0, stream>>>(rg_natoms, partials, offsets, G);

    for (int l = 0; l < L; ++l) {
        // --- GINEConv edge aggregation ---
        hipMemsetAsync(bufN1, 0, (size_t)N * H * 4, stream);
        edge_msg_kernel<<<E, H, 0, stream>>>(x, edge_feat, src, dst,
                                             bond_emb + (size_t)l * 3 * 100 * H, bufN1);
        // --- atom GIN MLP: h1 = ((1+eps)x + aggr) @ W1 + b1 ---
        gemm_wmma_kernel<<<dim3(N / 32, 2), 256, 0, stream>>>(
            x, bufN1, atom_eps + l, atom_W1 + (size_t)l * H * 2 * H,
            atom_b1 + (size_t)l * 2 * H, bufN2, N, H, 2 * H);
        hipMemsetAsync(stats, 0, 2 * 256 * 8, stream);
        bn_stats_kernel<<<dim3(2, gyN), H, 0, stream>>>(bufN2, N, 2 * H, RPB, stats);
        bn_relu_kernel<<<(unsigned)(((size_t)N * 2 * H + 255) / 256), 256, 0, stream>>>(
            bufN2, N, 2 * H, stats, atom_bn1g + (size_t)l * 2 * H, atom_bn1b + (size_t)l * 2 * H);
        gemm_wmma_kernel<<<dim3(N / 32, 1), 256, 0, stream>>>(
            bufN2, nullptr, nullptr, atom_W2 + (size_t)l * 2 * H * H,
            atom_b2 + (size_t)l * H, x, N, 2 * H, H);
        hipMemsetAsync(stats, 0, 2 * 256 * 8, stream);
        bn_stats_kernel<<<dim3(1, gyN), H, 0, stream>>>(x, N, H, RPB, stats);
        bn_relu_kernel<<<(unsigned)(((size_t)N * H + 255) / 256), 256, 0, stream>>>(
            x, N, H, stats, atom_bng + (size_t)l * H, atom_bnb + (size_t)l * H);

        // --- raw -> reduced graph:  rg = rg0 + relu(mean @ W + b) ---
        hipMemsetAsync(rgA, 0, (size_t)NR * H * 4, stream);
        hipMemsetAsync(countsR, 0, (size_t)NR * 4, stream);
        scatter_add_kernel<<<N, H, 0, stream>>>(x, mrow, mcol, rgA, countsR);
        div_counts_kernel<<<NR, H, 0, stream>>>(rgA, countsR);
        gemm_wmma_kernel<<<dim3(NR / 32, 1), 256, 0, stream>>>(
            rgA, nullptr, nullptr, raw2rg_W + (size_t)l * H * H,
            raw2rg_b + (size_t)l * H, rgB, NR, H, H);
        add_relu_kernel<<<(unsigned)(((size_t)NR * H + 255) / 256), 256, 0, stream>>>(
            rg0, rgB, rg, (size_t)NR * H);

        // --- reduced-graph GINConv ---
        hipMemsetAsync(rgA, 0, (size_t)NR * H * 4, stream);
        rg_aggr_kernel<<<ERG, H, 0, stream>>>(rg, rsrc, rdst, rgA);
        gemm_wmma_kernel<<<dim3(NR / 32, 2), 256, 0, stream>>>(
            rg, rgA, rg_eps + l, rg_W1 + (size_t)l * H * 2 * H,
            rg_b1 + (size_t)l * 2 * H, bufN2, NR, H, 2 * H);
        hipMemsetAsync(stats, 0, 2 * 256 * 8, stream);
        bn_stats_kernel<<<dim3(2, gyNR), H, 0, stream>>>(bufN2, NR, 2 * H, RPB, stats);
        bn_relu_kernel<<<(unsigned)(((size_t)NR * 2 * H + 255) / 256), 256, 0, stream>>>(
            bufN2, NR, 2 * H, stats, rg_bn1g + (size_t)l * 2 * H, rg_bn1b + (size_t)l * 2 * H);
        gemm_wmma_kernel<<<dim3(NR / 32, 1), 256, 0, stream>>>(
            bufN2, nullptr, nullptr, rg_W2 + (size_t)l * 2 * H * H,
            rg_b2 + (size_t)l * H, rg, NR, 2 * H, H);
        hipMemsetAsync(stats, 0, 2 * 256 * 8, stream);
        bn_stats_kernel<<<dim3(1, gyNR), H, 0, stream>>>(rg, NR, H, RPB, stats);
        bn_relu_kernel<<<(unsigned)(((size_t)NR * H + 255) / 256), 256, 0, stream>>>(
            rg, NR, H, stats, rg_bng + (size_t)l * H, rg_bnb + (size_t)l * H);

        // --- reduced -> raw:  x = x + relu(mean @ W + b) ---
        hipMemsetAsync(bufN1, 0, (size_t)N * H * 4, stream);
        hipMemsetAsync(countsN, 0, (size_t)N * 4, stream);
        scatter_add_kernel<<<N, H, 0, stream>>>(rg, mcol, mrow, bufN1, countsN);
        div_counts_kernel<<<N, H, 0, stream>>>(bufN1, countsN);
        gemm_wmma_kernel<<<dim3(N / 32, 1), 256, 0, stream>>>(
            bufN1, nullptr, nullptr, rg2raw_W + (size_t)l * H * H,
            rg2raw_b + (size_t)l * H, bufN2, N, H, H);
        add_relu_kernel<<<(unsigned)(((size_t)N * H + 255) / 256), 256, 0, stream>>>(
            x, bufN2, x, (size_t)N * H);
    }

    // ---- readout ----
    hipMemsetAsync(poolX, 0, (size_t)G * H * 4, stream);
    hipMemsetAsync(countsG, 0, (size_t)G * 4, stream);
    scatter_add_kernel<<<N, H, 0, stream>>>(x, nullptr, batch, poolX, countsG);
    div_counts_kernel<<<G, H, 0, stream>>>(poolX, countsG);
    gemm_wmma_kernel<<<dim3(G / 32, 1), 256, 0, stream>>>(
        poolX, nullptr, nullptr, a_lin_W, a_lin_b, xg, G, H, H);
    tree_pool_kernel<<<G, H, 0, stream>>>(rg0, offsets, rg_natoms, poolR);
    gemm_wmma_kernel<<<dim3(G / 32, 1), 256, 0, stream>>>(
        poolR, nullptr, nullptr, r_lin_W, r_lin_b, rgp, G, H, H);
    final_head_kernel<<<G, H, 0, stream>>>(xg, rgp, lin_W, lin_b, (float*)d_out);
}